// GraphSAGE_13975823581432
// MI455X (gfx1250) — compile-verified
//
#include <hip/hip_runtime.h>
#include <math.h>

#define NODES 80000

typedef float v2f __attribute__((ext_vector_type(2)));
typedef float v8f __attribute__((ext_vector_type(8)));

// ---------------------------------------------------------------------------
// degree: deg[dst] += 1 per edge
// ---------------------------------------------------------------------------
__global__ void deg_kernel(const int* __restrict__ dst, float* __restrict__ deg, int E) {
    int e = blockIdx.x * blockDim.x + threadIdx.x;
    if (e < E) atomicAdd(&deg[dst[e]], 1.0f);
}

// ---------------------------------------------------------------------------
// scatter-add projected features: acc[dst] += feat[src], F = 1<<logF wide
// ---------------------------------------------------------------------------
__global__ void scatter_kernel(const float* __restrict__ feat,
                               const int* __restrict__ src,
                               const int* __restrict__ dst,
                               float* __restrict__ acc,
                               int E, int logF) {
    long long gid = (long long)blockIdx.x * blockDim.x + threadIdx.x;
    int e = (int)(gid >> logF);
    if (e >= E) return;
    int f = (int)(gid & ((1 << logF) - 1));
    int s = src[e];
    int d = dst[e];
    atomicAdd(&acc[(size_t)d * (1 << logF) + f],
              feat[(size_t)s * (1 << logF) + f]);
}

// ---------------------------------------------------------------------------
// WMMA f32 GEMM: Y[M,N] = X[M,64] @ W[64,N] via V_WMMA_F32_16X16X4_F32.
// Each wave owns one 16-wide N strip; B fragments (the weights) are loaded
// ONCE into 16 v2f registers, then the wave marches over M tiles with a
// grid stride, so the hot loop is 16 x { global_load_b64 (imm offset) + wmma }.
// fuse==1: Y = elu(AGG/max(DEG,1) + X@W + BIAS)   (SAGE epilogue)
// ---------------------------------------------------------------------------
__global__ void wmma_gemm(const float* __restrict__ X,
                          const float* __restrict__ W,
                          const float* __restrict__ AGG,
                          const float* __restrict__ DEG,
                          const float* __restrict__ BIAS,
                          float* __restrict__ Y,
                          int M, int N, int fuse) {
    const int lane   = threadIdx.x & 31;
    const int wid    = blockIdx.x * (blockDim.x >> 5) + (threadIdx.x >> 5);
    const int nwaves = gridDim.x * (blockDim.x >> 5);

    const int ntiles = N >> 4;           // N-tiles (1 or 4)
    const int mtiles = M >> 4;           // 5000 row tiles
    const int nt       = wid % ntiles;   // this wave's N strip (fixed)
    const int mt0      = wid / ntiles;
    const int mtStride = nwaves / ntiles;   // launch guarantees divisibility

    const int r16   = lane & 15;         // A row / B col within tile
    const int khalf = (lane >> 4) << 1;  // K sub-pair: 0 (lanes 0-15) / 2 (16-31)
    const int col   = nt * 16 + r16;
    const int rbase = (lane >> 4) << 3;  // C rows: r (lanes 0-15) / r+8 (16-31)

    // ---- hoist B (weights) into registers: 16 k-steps x v2f ----
    v2f bfrag[16];
    {
        const float* Bcol = W + col;
#pragma unroll
        for (int kk = 0; kk < 16; ++kk) {
            int k = kk * 4 + khalf;
            bfrag[kk].x = Bcol[(size_t)k * N];
            bfrag[kk].y = Bcol[(size_t)(k + 1) * N];
        }
    }
    const float bias = fuse ? BIAS[col] : 0.0f;

    // ---- march over M tiles ----
    for (int mt = mt0; mt < mtiles; mt += mtStride) {
        const float* Arow = X + ((size_t)(mt * 16 + r16) * 64 + khalf);
        // prefetch next tile's A row into cache (gfx1250 global_prefetch_b8)
        __builtin_prefetch(Arow + (size_t)mtStride * 16 * 64, 0, 1);

        v8f c = {};
#pragma unroll
        for (int kk = 0; kk < 16; ++kk) {
            v2f a = *(const v2f*)(Arow + kk * 4);   // imm-offset b64 load
            c = __builtin_amdgcn_wmma_f32_16x16x4_f32(
                    /*neg_a=*/false, a, /*neg_b=*/false, bfrag[kk],
                    /*c_mod=*/(short)0, c, /*reuse_a=*/false, /*reuse_b=*/false);
        }

        const int rowBase = mt * 16 + rbase;
        float* Yp = Y + (size_t)rowBase * N + col;
        if (fuse) {
            const float* Ap = AGG + (size_t)rowBase * N + col;
            const float* Dp = DEG + rowBase;
#pragma unroll
            for (int r = 0; r < 8; ++r) {
                float inv = 1.0f / fmaxf(Dp[r], 1.0f);
                float v = Ap[(size_t)r * N] * inv + c[r] + bias;
                v = (v > 0.0f) ? v : expm1f(v);     // ELU (alpha = 1)
                Yp[(size_t)r * N] = v;
            }
        } else {
#pragma unroll
            for (int r = 0; r < 8; ++r) {
                Yp[(size_t)r * N] = c[r];
            }
        }
    }
}

// ---------------------------------------------------------------------------
// launch
// ---------------------------------------------------------------------------
extern "C" void kernel_launch(void* const* d_in, const int* in_sizes, int n_in,
                              void* d_out, int out_size, void* d_ws, size_t ws_size,
                              hipStream_t stream) {
    const float* x   = (const float*)d_in[0];
    const int*   ei  = (const int*)d_in[1];
    const float* Wl0 = (const float*)d_in[2];
    const float* Wr0 = (const float*)d_in[3];
    const float* b0  = (const float*)d_in[4];
    const float* Wl1 = (const float*)d_in[5];
    const float* Wr1 = (const float*)d_in[6];
    const float* b1  = (const float*)d_in[7];
    float*       out = (float*)d_out;

    const int E   = in_sizes[1] / 2;
    const int* src = ei;        // edge_index row 0
    const int* dst = ei + E;    // edge_index row 1

    // workspace layout (floats): deg | bufA (proj) | bufB (agg) | bufH (h)
    float* deg  = (float*)d_ws;
    float* bufA = deg  + NODES;
    float* bufB = bufA + (size_t)NODES * 64;
    float* bufH = bufB + (size_t)NODES * 64;

    hipMemsetAsync(deg,  0, (size_t)NODES * sizeof(float), stream);
    hipMemsetAsync(bufB, 0, (size_t)NODES * 64 * sizeof(float), stream);

    deg_kernel<<<(E + 255) / 256, 256, 0, stream>>>(dst, deg, E);

    // 512 blocks x 8 waves = 4096 persistent waves; 4096 % ntiles == 0 for
    // ntiles in {1, 4}, so mtStride is exact.
    const dim3 gBlocks(512), gThreads(256);

    // ---- layer 0 ----
    // xl = x @ Wl0   (project first; aggregation commutes with the linear map)
    wmma_gemm<<<gBlocks, gThreads, 0, stream>>>(
        x, Wl0, nullptr, nullptr, nullptr, bufA, NODES, 64, 0);
    // agg0[dst] += xl[src]  (64-wide)
    {
        long long th = (long long)E * 64;
        scatter_kernel<<<(int)((th + 255) / 256), 256, 0, stream>>>(
            bufA, src, dst, bufB, E, 6);
    }
    // h = elu(agg0/deg + x @ Wr0 + b0)
    wmma_gemm<<<gBlocks, gThreads, 0, stream>>>(
        x, Wr0, bufB, deg, b0, bufH, NODES, 64, 1);

    // ---- layer 1 ----
    // hl = h @ Wl1  (16-wide: 4x less scatter traffic than aggregating h)
    wmma_gemm<<<gBlocks, gThreads, 0, stream>>>(
        bufH, Wl1, nullptr, nullptr, nullptr, bufA, NODES, 16, 0);
    hipMemsetAsync(bufB, 0, (size_t)NODES * 16 * sizeof(float), stream);
    // agg1[dst] += hl[src]  (16-wide)
    {
        long long th = (long long)E * 16;
        scatter_kernel<<<(int)((th + 255) / 256), 256, 0, stream>>>(
            bufA, src, dst, bufB, E, 4);
    }
    // out = elu(agg1/deg + h @ Wr1 + b1)
    wmma_gemm<<<gBlocks, gThreads, 0, stream>>>(
        bufH, Wr1, bufB, deg, b1, out, NODES, 16, 1);
}